// DensityNet_56719338111406
// MI455X (gfx1250) — compile-verified
//
#include <hip/hip_runtime.h>

typedef float v2f __attribute__((ext_vector_type(2)));
typedef float v8f __attribute__((ext_vector_type(8)));

#define WAVES_PER_BLOCK 8
#define BLOCK_SIZE 256

__global__ __launch_bounds__(BLOCK_SIZE) void density_net_kernel(
    const float* __restrict__ X,   // [B,3]
    const float* __restrict__ W1,  // [3,32] row-major
    const float* __restrict__ b1,  // [32]
    const float* __restrict__ W2,  // [32,4] row-major
    const float* __restrict__ b2,  // [4]
    float* __restrict__ out,       // [B,8] (complex64 [B,2,2] flattened)
    int B)
{
    // wave-private LDS scratch: h tile (16x32) + params tile (16x16)
    __shared__ __align__(16) float lds_h[WAVES_PER_BLOCK][16 * 32];
    __shared__ __align__(16) float lds_p[WAVES_PER_BLOCK][16 * 16];

    const int lane = threadIdx.x & 31;
    const int wid  = threadIdx.x >> 5;
    const int col  = lane & 15;   // N (B/C/D) or M (A) index
    const int hi   = lane >> 4;   // lane half: selects K pair

    float* hbuf = lds_h[wid];
    float* pbuf = lds_p[wid];

    // ---------- loop-invariant B fragments ----------
    // Layer 1: B = W1aug (4x32), row 3 = b1.  Fragment h covers cols 16h+col,
    // rows {2*hi, 2*hi+1}.
    v2f B1f[2];
#pragma unroll
    for (int h = 0; h < 2; ++h) {
        int n = col + 16 * h;
        B1f[h].x = W1[(2 * hi) * 32 + n];            // row 0 or row 2 of W1
        B1f[h].y = hi ? b1[n] : W1[1 * 32 + n];      // row 3 = bias, else row 1
    }

    // Layer 2: 8 chained K=4 slabs of W2 (32x4). Columns >=4 are don't-care -> 0.
    v2f B2f[8];
#pragma unroll
    for (int k = 0; k < 8; ++k) {
        int r0 = 4 * k + 2 * hi;
        B2f[k].x = (col < 4) ? W2[r0 * 4 + col]       : 0.0f;
        B2f[k].y = (col < 4) ? W2[(r0 + 1) * 4 + col] : 0.0f;
    }

    // bias b2 broadcast into C (same value in all 8 C VGPRs: column-only dependence)
    const float biasc = (col < 4) ? b2[col] : 0.0f;
    v8f Cbias;
#pragma unroll
    for (int i = 0; i < 8; ++i) Cbias[i] = biasc;

    const v8f zero8 = {0.f, 0.f, 0.f, 0.f, 0.f, 0.f, 0.f, 0.f};

    const int ntiles  = (B + 15) >> 4;
    const int nwaves  = (gridDim.x * BLOCK_SIZE) >> 5;
    const int wave_id = (blockIdx.x * BLOCK_SIZE + threadIdx.x) >> 5;

    for (int tile = wave_id; tile < ntiles; tile += nwaves) {
        const int m0 = tile << 4;

        // ---- A fragment for layer 1: row = col, K pair per lane half ----
        // Uniform (branch-free) load: hi=0 reads {x0,x1}, hi=1 reads {x1,x2},
        // then cndmask-select.  Max offset row*3+2 -> never OOB.
        int row  = m0 + col;
        int rowc = row < B ? row : (B - 1);
        const float* xr = X + (size_t)rowc * 3;
        float a0 = xr[hi];
        float a1 = xr[hi + 1];
        v2f A1;
        A1.x = hi ? a1 : a0;     // K=0: x0   | K=2: x2
        A1.y = hi ? 1.0f : a1;   // K=1: x1   | K=3: 1.0 (bias row)

        // ---- layer 1: h = relu(Xaug @ W1aug) ----
        v8f H0 = __builtin_amdgcn_wmma_f32_16x16x4_f32(false, A1, false, B1f[0],
                                                       (short)0, zero8, false, false);
        v8f H1 = __builtin_amdgcn_wmma_f32_16x16x4_f32(false, A1, false, B1f[1],
                                                       (short)0, zero8, false, false);
#pragma unroll
        for (int i = 0; i < 8; ++i) {
            H0[i] = fmaxf(H0[i], 0.0f);
            H1[i] = fmaxf(H1[i], 0.0f);
        }

        // ---- C-layout -> row-major LDS (16x32) ----
#pragma unroll
        for (int r = 0; r < 8; ++r) {
            int m = r + 8 * hi;
            hbuf[m * 32 + col]      = H0[r];
            hbuf[m * 32 + col + 16] = H1[r];
        }
        asm volatile("s_wait_dscnt 0" ::: "memory");  // wave-private; DS is in-order

        // ---- layer 2: P = h @ W2 + b2, 8 chained K=4 WMMAs ----
        v8f P = Cbias;
#pragma unroll
        for (int k = 0; k < 8; ++k) {
            v2f A2 = *(const v2f*)(hbuf + col * 32 + 4 * k + 2 * hi);  // 8B-aligned
            P = __builtin_amdgcn_wmma_f32_16x16x4_f32(false, A2, false, B2f[k],
                                                      (short)0, P, false, false);
        }

        // ---- P fragment -> LDS (full 16x16; only cols 0..3 meaningful) ----
#pragma unroll
        for (int r = 0; r < 8; ++r) {
            pbuf[(r + 8 * hi) * 16 + col] = P[r];
        }
        asm volatile("s_wait_dscnt 0" ::: "memory");

        // ---- epilogue: rho = L L^H / tr, lane = (row=col, half=hi) ----
        {
            const float* pr = pbuf + col * 16;
            float p0 = pr[0], p1 = pr[1], p2 = pr[2], p3 = pr[3];
            float r00 = p0 * p0;
            float r11 = p1 * p1 + p2 * p2 + p3 * p3;
            float inv = 1.0f / (r00 + r11);
            float a  = p0 * p1;
            float bb = p0 * p2;
            float4 o;
            if (hi) o = make_float4(a * inv,   bb * inv, r11 * inv, 0.0f);     // rho10, rho11
            else    o = make_float4(r00 * inv, 0.0f,     a * inv,  -bb * inv); // rho00, rho01
            int orow = m0 + col;
            if (orow < B)
                *(float4*)(out + (size_t)orow * 8 + 4 * hi) = o;
        }
    }
}

extern "C" void kernel_launch(void* const* d_in, const int* in_sizes, int n_in,
                              void* d_out, int out_size, void* d_ws, size_t ws_size,
                              hipStream_t stream) {
    const float* X  = (const float*)d_in[0];
    const float* W1 = (const float*)d_in[1];
    const float* b1 = (const float*)d_in[2];
    const float* W2 = (const float*)d_in[3];
    const float* b2 = (const float*)d_in[4];
    float* out = (float*)d_out;

    const int B = in_sizes[0] / 3;
    const int ntiles = (B + 15) / 16;

    int blocks = 2048;  // 16384 waves -> 8 tiles/wave at B = 2M
    int max_blocks = (ntiles + WAVES_PER_BLOCK - 1) / WAVES_PER_BLOCK;
    if (blocks > max_blocks) blocks = max_blocks;
    if (blocks < 1) blocks = 1;

    density_net_kernel<<<blocks, BLOCK_SIZE, 0, stream>>>(X, W1, b1, W2, b2, out, B);
}